// GRUD_52879637348603
// MI455X (gfx1250) — compile-verified
//
#include <hip/hip_runtime.h>
#include <stdint.h>

// ---------------- problem geometry ----------------
#define F_TOTAL 4096
#define T_TOTAL 2000
#define TCHUNK  8
#define NCHUNK  (T_TOTAL / TCHUNK)      // 250
#define BLOCK   256
#define ROWPAD  (TCHUNK + 1)            // 9 dwords per row in LDS (bank-conflict-free)
#define CH_DWORDS (BLOCK * ROWPAD)      // 2304 dwords per channel tile
#define CH_BYTES  (CH_DWORDS * 4)       // 9216 B
#define BUF_BYTES (3 * CH_BYTES)        // 27648 B per double-buffer half

typedef uint32_t u32x4 __attribute__((ext_vector_type(4)));
typedef uint32_t u32x8 __attribute__((ext_vector_type(8)));

#define LOG2E 1.4426950408889634f

__device__ __forceinline__ float fexp2(float x) { return __builtin_amdgcn_exp2f(x); }
__device__ __forceinline__ float frcp(float x)  { return __builtin_amdgcn_rcpf(x); }
__device__ __forceinline__ float fsigmoid(float x) {
  return frcp(1.0f + fexp2(-LOG2E * x));           // v_exp + v_add + v_rcp
}
__device__ __forceinline__ float ftanh(float x) {
  // tanh(x) = 1 - 2/(exp(2x)+1)
  return fmaf(-2.0f, frcp(fexp2((2.0f * LOG2E) * x) + 1.0f), 1.0f);
}

// Issue one TDM 2D tile load: tile = [BLOCK rows x TCHUNK dwords] out of a
// [F_TOTAL x T_TOTAL] f32 matrix whose row stride is T_TOTAL dwords.
// Pads +1 dword after every 8 dwords stored -> 9-dword LDS row stride.
__device__ __forceinline__ void tdm_load_tile(uint64_t gaddr, uint32_t lds_off) {
  u32x4 g0;
  g0[0] = 1u;                                        // count=1, user descriptor
  g0[1] = lds_off;                                   // LDS byte address
  g0[2] = (uint32_t)gaddr;                           // global_addr[31:0]
  g0[3] = ((uint32_t)(gaddr >> 32) & 0x01FFFFFFu)    // global_addr[56:32]
        | (2u << 30);                                // type = 2 ("image")
  u32x8 g1;
  g1[0] = (2u << 16)        // data_size = 4 bytes
        | (1u << 20)        // pad_enable
        | (2u << 22)        // pad_interval: 8 dwords
        | (0u << 25);       // pad_amount: 1 dword
  g1[1] = ((uint32_t)T_TOTAL) << 16;                 // tensor_dim0[15:0] @ bits 63:48
  g1[2] = ((uint32_t)F_TOTAL) << 16;                 // tensor_dim0[31:16]=0 | tensor_dim1[15:0]
  g1[3] = ((uint32_t)TCHUNK) << 16;                  // tensor_dim1[31:16]=0 | tile_dim0
  g1[4] = (uint32_t)BLOCK;                           // tile_dim1 | tile_dim2=0
  g1[5] = (uint32_t)T_TOTAL;                         // tensor_dim0_stride[31:0]
  g1[6] = 0u;                                        // stride0[47:32] | stride1[15:0]
  g1[7] = 0u;                                        // stride1[47:16]
  asm volatile("tensor_load_to_lds %0, %1" :: "s"(g0), "s"(g1) : "memory");
}

__global__ __launch_bounds__(BLOCK) void grud_scan(
    const float* __restrict__ inp,
    const float* __restrict__ xmean_p,
    const float* __restrict__ wdgx_p, const float* __restrict__ wdgh_p,
    const float* __restrict__ wxz_p,  const float* __restrict__ whz_p,
    const float* __restrict__ wmz_p,
    const float* __restrict__ wxr_p,  const float* __restrict__ whr_p,
    const float* __restrict__ wmr_p,
    const float* __restrict__ wxh_p,  const float* __restrict__ whh_p,
    const float* __restrict__ wmh_p,
    const float* __restrict__ bdgx_p, const float* __restrict__ bdgh_p,
    const float* __restrict__ bz_p,   const float* __restrict__ br_p,
    const float* __restrict__ bh_p,
    float* __restrict__ hout)
{
  __shared__ float lds[2 * 3 * CH_DWORDS];           // 55296 B, double buffered

  const int tid = threadIdx.x;
  const int f0  = blockIdx.x * BLOCK;
  const int f   = f0 + tid;

  // Per-element weights live in registers for the whole scan.
  const float wdgx = wdgx_p[f], bdgx = bdgx_p[f];
  const float wdgh = wdgh_p[f], bdgh = bdgh_p[f];
  const float wxz = wxz_p[f], whz = whz_p[f], wmz = wmz_p[f], bz = bz_p[f];
  const float wxr = wxr_p[f], whr = whr_p[f], wmr = wmr_p[f], br = br_p[f];
  const float wxh = wxh_p[f], whh = whh_p[f], wmh = wmh_p[f], bh = bh_p[f];
  const float xmean = xmean_p[f];

  const uint32_t lds_base = (uint32_t)(uintptr_t)(&lds[0]); // low 32 bits = LDS offset
  const uint64_t gbase    = (uint64_t)(uintptr_t)inp;

  // Scalar (wave-uniform) condition -> s_cbranch, so exactly one wave per
  // block executes the TDM issues (TDM ignores EXEC).
  const bool issuer = (__builtin_amdgcn_readfirstlane((int)tid) == 0);

  // Prologue: chunk 0 -> buffer 0 (3 channels: X, M, D)
  if (issuer) {
#pragma unroll
    for (int c = 0; c < 3; ++c)
      tdm_load_tile(gbase + (uint64_t)(c * F_TOTAL + f0) * (uint64_t)T_TOTAL * 4ull,
                    lds_base + (uint32_t)c * CH_BYTES);
  }

  float h = 0.0f;

  for (int k = 0; k < NCHUNK; ++k) {
    __builtin_amdgcn_s_wait_tensorcnt(0);  // chunk k landed (no-op for non-issuer waves)
    __syncthreads();                       // publish LDS tile; also fences last chunk's readers

    if ((k + 1 < NCHUNK) && issuer) {      // stream chunk k+1 while computing chunk k
      const uint64_t t1 = (uint64_t)((k + 1) * TCHUNK);
      const uint32_t b1 = (uint32_t)((k + 1) & 1) * (uint32_t)BUF_BYTES;
#pragma unroll
      for (int c = 0; c < 3; ++c)
        tdm_load_tile(gbase + ((uint64_t)(c * F_TOTAL + f0) * (uint64_t)T_TOTAL + t1) * 4ull,
                      lds_base + b1 + (uint32_t)c * CH_BYTES);
    }

    const float* bufX = &lds[(k & 1) * 3 * CH_DWORDS + tid * ROWPAD];
    const float* bufM = bufX + CH_DWORDS;
    const float* bufD = bufM + CH_DWORDS;

#pragma unroll
    for (int tr = 0; tr < TCHUNK; ++tr) {
      const float x = bufX[tr];
      const float m = bufM[tr];
      const float d = bufD[tr];

      // Everything below until `h *= gh` is independent of h -> off the chain.
      const float gx = fexp2(-LOG2E * fmaxf(0.0f, fmaf(wdgx, d, bdgx)));
      const float gh = fexp2(-LOG2E * fmaxf(0.0f, fmaf(wdgh, d, bdgh)));
      const float xp = m * x + (1.0f - m) * fmaf(gx, x, (1.0f - gx) * xmean);
      const float az = fmaf(wxz, xp, fmaf(wmz, m, bz));
      const float ar = fmaf(wxr, xp, fmaf(wmr, m, br));
      const float ah = fmaf(wxh, xp, fmaf(wmh, m, bh));

      h *= gh;
      const float z  = fsigmoid(fmaf(whz, h, az));
      const float r  = fsigmoid(fmaf(whr, h, ar));
      const float ht = ftanh(fmaf(whh, r * h, ah));
      h = fmaf(z, ht - h, h);
    }
  }

  hout[f] = h;
}

// Output head: y[o] = sum_f w_hy[o,f] * h[f] + b_y[o], o in {0,1}
__global__ __launch_bounds__(BLOCK) void grud_head(
    const float* __restrict__ h, const float* __restrict__ why,
    const float* __restrict__ by, float* __restrict__ out)
{
  __shared__ float red0[BLOCK];
  __shared__ float red1[BLOCK];
  const int tid = threadIdx.x;
  float s0 = 0.0f, s1 = 0.0f;
  for (int f = tid; f < F_TOTAL; f += BLOCK) {
    const float hv = h[f];
    s0 = fmaf(why[f], hv, s0);
    s1 = fmaf(why[F_TOTAL + f], hv, s1);
  }
  red0[tid] = s0;
  red1[tid] = s1;
  __syncthreads();
  for (int off = BLOCK / 2; off > 0; off >>= 1) {
    if (tid < off) {
      red0[tid] += red0[tid + off];
      red1[tid] += red1[tid + off];
    }
    __syncthreads();
  }
  if (tid == 0) {
    out[0] = red0[0] + by[0];
    out[1] = red1[0] + by[1];
  }
}

extern "C" void kernel_launch(void* const* d_in, const int* in_sizes, int n_in,
                              void* d_out, int out_size, void* d_ws, size_t ws_size,
                              hipStream_t stream) {
  const float* inp   = (const float*)d_in[0];
  const float* xmean = (const float*)d_in[1];
  const float* wdgx  = (const float*)d_in[2];
  const float* wdgh  = (const float*)d_in[3];
  const float* wxz   = (const float*)d_in[4];
  const float* whz   = (const float*)d_in[5];
  const float* wmz   = (const float*)d_in[6];
  const float* wxr   = (const float*)d_in[7];
  const float* whr   = (const float*)d_in[8];
  const float* wmr   = (const float*)d_in[9];
  const float* wxh   = (const float*)d_in[10];
  const float* whh   = (const float*)d_in[11];
  const float* wmh   = (const float*)d_in[12];
  const float* why   = (const float*)d_in[13];
  const float* bdgx  = (const float*)d_in[14];
  const float* bdgh  = (const float*)d_in[15];
  const float* bz    = (const float*)d_in[16];
  const float* br    = (const float*)d_in[17];
  const float* bh    = (const float*)d_in[18];
  const float* by    = (const float*)d_in[19];

  float* h = (float*)d_ws;  // 4096 floats of scratch for the final hidden state

  grud_scan<<<F_TOTAL / BLOCK, BLOCK, 0, stream>>>(
      inp, xmean, wdgx, wdgh, wxz, whz, wmz, wxr, whr, wmr,
      wxh, whh, wmh, bdgx, bdgh, bz, br, bh, h);
  grud_head<<<1, BLOCK, 0, stream>>>(h, why, by, (float*)d_out);
}